// EcholancerLoss_62405874811003
// MI455X (gfx1250) — compile-verified
//
#include <hip/hip_runtime.h>
#include <math.h>
#include <stdint.h>

// ---- problem constants (match reference) ----
#define V_TEXT_   256
#define V_AUDIO_  4096
#define V_TOTAL_  4352
#define B_        8
#define H_        4
#define TQ_       800
#define TK_       128
#define TTOK_     1024
#define NROWS_    (B_ * TTOK_)        // 8192 CE rows
#define NSEQ_     (B_ * H_)           // 32 CTC sequences
#define SPL_      9                   // states per lane (9*32=288 >= 257)
#define S_        257                 // 2*TK+1
#define NEGC_     (-1000000000.0f)
#define BLANK_LP_ (-8.0f)
#define CE_W_     1.5f
#define ATTN_W_   10.0f
#define ATTN_START_ 5000

typedef float v2f_ __attribute__((ext_vector_type(2)));
typedef float v8f_ __attribute__((ext_vector_type(8)));

// =====================================================================
// Kernel 1: masked cross-entropy rows. One 256-thread block per (b,t).
// Writes per-row NLL (0 if invalid) to ws_nll[row].
// =====================================================================
__global__ __launch_bounds__(256)
void ech_ce_kernel(const float* __restrict__ logits,
                   const int* __restrict__ targets,
                   const int* __restrict__ alens,
                   float* __restrict__ nll_out)
{
    const int row = blockIdx.x;               // 0..8191
    const int b   = row >> 10;                // /1024
    const int t   = row & (TTOK_ - 1);
    const float* __restrict__ r = logits + (size_t)row * V_TOTAL_ + V_TEXT_;
    const float4* __restrict__ r4 = (const float4*)r;   // 16B-aligned (1024B offset)
    const int tid  = threadIdx.x;
    const int lane = tid & 31;
    const int wid  = tid >> 5;

    __shared__ float redA[8];
    __shared__ float redB[8];

    // 16 values per thread, kept in registers for both passes
    float4 v[4];
#pragma unroll
    for (int j = 0; j < 4; ++j) v[j] = r4[tid + 256 * j];

    float mx = -3.4e38f;
#pragma unroll
    for (int j = 0; j < 4; ++j) {
        mx = fmaxf(mx, fmaxf(fmaxf(v[j].x, v[j].y), fmaxf(v[j].z, v[j].w)));
    }
#pragma unroll
    for (int off = 16; off; off >>= 1) mx = fmaxf(mx, __shfl_xor(mx, off, 32));
    if (lane == 0) redA[wid] = mx;
    __syncthreads();
    float M = redA[0];
#pragma unroll
    for (int j = 1; j < 8; ++j) M = fmaxf(M, redA[j]);

    float s = 0.0f;
#pragma unroll
    for (int j = 0; j < 4; ++j) {
        s += __expf(v[j].x - M) + __expf(v[j].y - M) +
             __expf(v[j].z - M) + __expf(v[j].w - M);
    }
#pragma unroll
    for (int off = 16; off; off >>= 1) s += __shfl_xor(s, off, 32);
    if (lane == 0) redB[wid] = s;
    __syncthreads();

    if (tid == 0) {
        float S = 0.0f;
#pragma unroll
        for (int j = 0; j < 8; ++j) S += redB[j];
        const float lse = M + __logf(S);
        const int tg = targets[row];
        int vt = tg - V_TEXT_;
        vt = vt < 0 ? 0 : (vt > V_AUDIO_ - 1 ? V_AUDIO_ - 1 : vt);
        const float nll = lse - r[vt];
        const bool valid = (t < alens[b]) && (tg != -100);
        nll_out[row] = valid ? nll : 0.0f;
    }
}

// =====================================================================
// Kernel 2: CTC forward-sum. One wave (32 lanes) per sequence, 9 states
// per lane. Attention rows are streamed global->LDS with CDNA5 async
// loads (quad-buffered, s_wait_asynccnt pipelining).
// =====================================================================
__device__ __forceinline__ void ech_async_row(const float* gaddr, uint32_t lds_byte)
{
    // Per ISA 15.18: the LDS and memory addresses take the same offset
    // -> one address pair, inst_offset advances both sides together.
    asm volatile(
        "global_load_async_to_lds_b32 %0, %1, off\n\t"
        "global_load_async_to_lds_b32 %0, %1, off offset:128\n\t"
        "global_load_async_to_lds_b32 %0, %1, off offset:256\n\t"
        "global_load_async_to_lds_b32 %0, %1, off offset:384"
        :: "v"(lds_byte), "v"(gaddr) : "memory");
}

__global__ __launch_bounds__(32)
void ech_attn_kernel(const float* __restrict__ attn,
                     const int* __restrict__ src_lens,
                     const int* __restrict__ out_lens,
                     float* __restrict__ loss_out)
{
    __shared__ float sm[4 * TK_];             // 4 row buffers of 128 floats
    const int n    = blockIdx.x;              // 0..31  (n = b*H + h)
    const int bb   = n / H_;
    const int lane = threadIdx.x;

    int k = src_lens[bb]; if (k > TK_) k = TK_;
    int q = out_lens[bb]; if (q > TQ_) q = TQ_;
    const int smax = 2 * k;                   // valid states: s <= 2k

    const float* __restrict__ base = attn + (size_t)n * TQ_ * TK_;
    const uint32_t lds_base = (uint32_t)(uintptr_t)(&sm[0]);

    // prefetch rows 0,1,2 (q >= 1 always; rows clamp inside tensor)
#pragma unroll
    for (int r0 = 0; r0 < 3; ++r0) {
        int tr = r0 < q ? r0 : (q - 1);
        ech_async_row(base + (size_t)tr * TK_ + lane,
                      lds_base + (uint32_t)((r0 & 3) * TK_ + lane) * 4u);
    }

    float a[SPL_];
#pragma unroll
    for (int i = 0; i < SPL_; ++i) a[i] = NEGC_;

    for (int t = 0; t < q; ++t) {
        // rows t,t+1,t+2 in flight (12 async ops); row t done when <= 8 remain
        asm volatile("s_wait_asynccnt 0x8" ::: "memory");
        {   // keep the pipe full: issue row t+3 into buffer (t+3)&3.
            // That buffer's ds_loads were issued at iteration t-1; drain the
            // LDS pipe first so the async writes can't pass pending reads.
            asm volatile("s_wait_dscnt 0x0" ::: "memory");
            int tn = t + 3; if (tn > TQ_ - 1) tn = TQ_ - 1;
            ech_async_row(base + (size_t)tn * TK_ + lane,
                          lds_base + (uint32_t)(((t + 3) & 3) * TK_ + lane) * 4u);
        }
        const float* __restrict__ bufp = &sm[(t & 3) * TK_];

        // ---- masked log-softmax over 129 classes (blank + 128) ----
        const float x0 = bufp[lane], x1 = bufp[lane + 32],
                    x2 = bufp[lane + 64], x3 = bufp[lane + 96];
        const bool v0 = (lane + 1)  <= k, v1 = (lane + 33) <= k,
                   v2 = (lane + 65) <= k, v3 = (lane + 97) <= k;
        const float y0 = v0 ? x0 : NEGC_, y1 = v1 ? x1 : NEGC_,
                    y2 = v2 ? x2 : NEGC_, y3 = v3 ? x3 : NEGC_;
        float mx = fmaxf(fmaxf(fmaxf(y0, y1), fmaxf(y2, y3)), BLANK_LP_);
#pragma unroll
        for (int off = 16; off; off >>= 1) mx = fmaxf(mx, __shfl_xor(mx, off, 32));
        float se = (v0 ? __expf(y0 - mx) : 0.0f) + (v1 ? __expf(y1 - mx) : 0.0f) +
                   (v2 ? __expf(y2 - mx) : 0.0f) + (v3 ? __expf(y3 - mx) : 0.0f);
#pragma unroll
        for (int off = 16; off; off >>= 1) se += __shfl_xor(se, off, 32);
        se += __expf(BLANK_LP_ - mx);         // blank class (always valid)
        const float lse = mx + __logf(se);

        // ---- emit values for this lane's 9 states ----
        float e[SPL_];
#pragma unroll
        for (int i = 0; i < SPL_; ++i) {
            const int s = SPL_ * lane + i;
            if (s & 1) {
                int bi = (s - 1) >> 1;        // label-1 in [0,127] for valid s
                if (bi > TK_ - 1) bi = TK_ - 1;   // s>=257 is masked anyway
                e[i] = bufp[bi] - lse;
            } else {
                e[i] = BLANK_LP_ - lse;
            }
        }

        if (t == 0) {
#pragma unroll
            for (int i = 0; i < SPL_; ++i) {
                const int s = SPL_ * lane + i;
                const float ini = (s < 2) ? e[i] : NEGC_;
                a[i] = (s <= smax) ? ini : NEGC_;
            }
        } else {
            float up1 = __shfl_up(a[SPL_ - 1], 1, 32);   // alpha[9L-1]
            float up2 = __shfl_up(a[SPL_ - 2], 1, 32);   // alpha[9L-2]
            if (lane == 0) { up1 = NEGC_; up2 = NEGC_; }
            float na[SPL_];
#pragma unroll
            for (int i = 0; i < SPL_; ++i) {
                const int s  = SPL_ * lane + i;
                const float c0 = a[i];
                const float c1 = (i >= 1) ? a[i - 1] : up1;
                float       c2 = (i >= 2) ? a[i - 2] : ((i == 1) ? up1 : up2);
                if (!((s & 1) && (s >= 2))) c2 = NEGC_;   // skip_ok mask
                const float m3 = fmaxf(fmaxf(c0, c1), c2);
                const float s3 = __expf(c0 - m3) + __expf(c1 - m3) + __expf(c2 - m3);
                const float nv = m3 + __logf(s3) + e[i];
                na[i] = (s <= smax) ? nv : NEGC_;
            }
#pragma unroll
            for (int i = 0; i < SPL_; ++i) a[i] = na[i];
        }
    }

    // ---- gather alpha[2k-1], alpha[2k]; -logaddexp; normalize ----
    const int s1 = 2 * k - 1, s2 = 2 * k;
    float g1 = NEGC_, g2 = NEGC_;
#pragma unroll
    for (int i = 0; i < SPL_; ++i) {
        const int s = SPL_ * lane + i;
        if (s == s1) g1 = a[i];
        if (s == s2) g2 = a[i];
    }
#pragma unroll
    for (int off = 16; off; off >>= 1) {
        g1 = fmaxf(g1, __shfl_xor(g1, off, 32));
        g2 = fmaxf(g2, __shfl_xor(g2, off, 32));
    }
    const float mm  = fmaxf(g1, g2);
    const float lae = mm + __logf(__expf(g1 - mm) + __expf(g2 - mm));
    float loss = (-lae) / (float)k;
    if (!(isfinite(loss) && loss < 1.0e8f)) loss = 0.0f;
    if (lane == 0) loss_out[n] = loss;
}

// =====================================================================
// Kernel 3: deterministic final reduction + gating + output.
// One wave; the 8192-element NLL sum is finished with a single
// V_WMMA_F32_16X16X4_F32 against an all-ones B (D[m,n] = sum_k A[m,k]).
// Summing all 256 D values counts every A element exactly 16 times, so
// total = sum(D)/16 independent of the exact A/B/D lane layouts.
// =====================================================================
__global__ __launch_bounds__(32)
void ech_final_kernel(const float* __restrict__ nll,
                      const float* __restrict__ attn_losses,
                      const int* __restrict__ alens,
                      const int* __restrict__ step,
                      float* __restrict__ out)
{
    const int lane = threadIdx.x;

    // 64 deterministic partial sums (slot s accumulates nll[s + 64*j])
    float p0 = 0.0f, p1 = 0.0f;
    for (int j = 0; j < NROWS_ / 64; ++j) {
        p0 += nll[lane      + 64 * j];
        p1 += nll[lane + 32 + 64 * j];
    }

    float token_sum;
#if __has_builtin(__builtin_amdgcn_wmma_f32_16x16x4_f32)
    {
        v2f_ A;  A.x = p0;  A.y = p1;        // 16x4 f32 A operand (2 VGPRs/wave)
        v2f_ Bv; Bv.x = 1.0f; Bv.y = 1.0f;   // 4x16 all-ones B
        v8f_ C = {};
        C = __builtin_amdgcn_wmma_f32_16x16x4_f32(
                /*neg_a=*/false, A, /*neg_b=*/false, Bv,
                /*c_mod=*/(short)0, C, /*reuse_a=*/false, /*reuse_b=*/false);
        float acc = C[0] + C[1] + C[2] + C[3] + C[4] + C[5] + C[6] + C[7];
#pragma unroll
        for (int off = 16; off; off >>= 1) acc += __shfl_xor(acc, off, 32);
        token_sum = acc * (1.0f / 16.0f);    // each row-sum replicated over 16 cols
    }
#else
    {
        float acc = p0 + p1;
#pragma unroll
        for (int off = 16; off; off >>= 1) acc += __shfl_xor(acc, off, 32);
        token_sum = acc;
    }
#endif

    if (lane == 0) {
        long long denom = 0;
#pragma unroll
        for (int b = 0; b < B_; ++b) {
            int c = alens[b];
            c = c < 0 ? 0 : (c > TTOK_ ? TTOK_ : c);
            denom += c;
        }
        if (denom < 1) denom = 1;
        const float token_loss = token_sum / (float)denom;

        float as = 0.0f;
#pragma unroll
        for (int i = 0; i < NSEQ_; ++i) as += attn_losses[i];
        float attn_loss = as / (float)NSEQ_;
        if (!(step[0] > ATTN_START_)) attn_loss = 0.0f;

        out[0] = token_loss * CE_W_ + attn_loss * ATTN_W_;
        out[1] = attn_loss;
        out[2] = token_loss;
    }
}

// =====================================================================
extern "C" void kernel_launch(void* const* d_in, const int* in_sizes, int n_in,
                              void* d_out, int out_size, void* d_ws, size_t ws_size,
                              hipStream_t stream)
{
    (void)in_sizes; (void)n_in; (void)out_size; (void)ws_size;
    const float* logits   = (const float*)d_in[0];   // [8,1024,4352] f32
    const float* attn     = (const float*)d_in[1];   // [8,4,800,128] f32
    const int*   targets  = (const int*)d_in[2];     // [8,1024] i32
    const int*   alens    = (const int*)d_in[3];     // [8] i32
    const int*   src_lens = (const int*)d_in[4];     // [8] i32
    const int*   out_lens = (const int*)d_in[5];     // [8] i32
    const int*   step     = (const int*)d_in[6];     // [1] i32
    float* outp = (float*)d_out;                     // [3] f32

    float* wsf        = (float*)d_ws;
    float* ws_nll     = wsf;                         // 8192 floats
    float* ws_attn    = wsf + NROWS_;                // 32 floats

    ech_ce_kernel<<<NROWS_, 256, 0, stream>>>(logits, targets, alens, ws_nll);
    ech_attn_kernel<<<NSEQ_, 32, 0, stream>>>(attn, src_lens, out_lens, ws_attn);
    ech_final_kernel<<<1, 32, 0, stream>>>(ws_nll, ws_attn, alens, step, outp);
}